// SpinAggregateLayer_86028194939355
// MI455X (gfx1250) — compile-verified
//
#include <hip/hip_runtime.h>

typedef __attribute__((ext_vector_type(2))) float v2f;
typedef __attribute__((ext_vector_type(8))) float v8f;

#define N_CHANNEL 16
#define CUTOFF    8.0f
#define INV_NORM  (1.0f/32.0f)
#define PI_F      3.14159265358979323846f

#define EDGES_PER_BLOCK 128
#define THREADS 256

__global__ void zero_f32_kernel(float* __restrict__ p, int n) {
    int i = blockIdx.x * blockDim.x + threadIdx.x;
    if (i < n) p[i] = 0.0f;
}

__global__ __launch_bounds__(THREADS)
void spin_edge_kernel(const float* __restrict__ tin,
                      const float* __restrict__ coord,
                      const float* __restrict__ W0,
                      const float* __restrict__ W1,
                      const float* __restrict__ W2,
                      const int*   __restrict__ idx_i,
                      const int*   __restrict__ idx_j,
                      float*       __restrict__ out,
                      int n_edges)
{
    __shared__ int   s_ii[EDGES_PER_BLOCK];
    __shared__ int   s_jj[EDGES_PER_BLOCK];
    __shared__ float s_ux[EDGES_PER_BLOCK];
    __shared__ float s_uy[EDGES_PER_BLOCK];
    __shared__ float s_uz[EDGES_PER_BLOCK];
    __shared__ float s_d [EDGES_PER_BLOCK];

    const int tid      = threadIdx.x;
    const int block_e0 = blockIdx.x * EDGES_PER_BLOCK;

    // ---- Stage 1: per-edge geometry into LDS (clamped for tail; EXEC stays full later)
    if (tid < EDGES_PER_BLOCK) {
        int e  = block_e0 + tid;
        int ec = (e < n_edges) ? e : (n_edges - 1);
        int ii = idx_i[ec];
        int jj = idx_j[ec];
        float xi = coord[ii*3+0], yi = coord[ii*3+1], zi = coord[ii*3+2];
        float xj = coord[jj*3+0], yj = coord[jj*3+1], zj = coord[jj*3+2];
        float rx = xj - xi, ry = yj - yi, rz = zj - zi;
        float d   = sqrtf(rx*rx + ry*ry + rz*rz);
        float inv = 1.0f / d;
        s_ii[tid] = ii;  s_jj[tid] = jj;
        s_ux[tid] = rx*inv; s_uy[tid] = ry*inv; s_uz[tid] = rz*inv;
        s_d [tid] = d;
    }
    __syncthreads();

    const int wave  = tid >> 5;
    const int lane  = tid & 31;
    const int h     = lane >> 4;   // half-wave
    const int m     = lane & 15;   // row (edge) for A, column (channel) for B/D
    const int wbase = wave * 16;   // this wave's 16-edge tile in LDS

    // ---- Stage 2: Bessel RBF -> A operands (16x4 f32 tiles, K split into 4 chunks)
    float d     = s_d[wbase + m];
    float invd  = 1.0f / d;
    float fc    = 0.5f * (__cosf((PI_F/CUTOFF) * fminf(d, CUTOFF)) + 1.0f);
    float scale = sqrtf(2.0f / CUTOFF) * invd * fc;
    float t     = (PI_F / CUTOFF) * d;

    v2f a[4];
    #pragma unroll
    for (int c = 0; c < 4; ++c) {
        float n0 = (float)(4*c + 2*h + 1);        // Bessel order n = k+1
        a[c].x = scale * __sinf(n0 * t);
        a[c].y = scale * __sinf((n0 + 1.0f) * t);
    }

    // B operands: W[k][n], row-striped across lanes (lane = channel n)
    v2f b0[4], b1[4], b2[4];
    #pragma unroll
    for (int c = 0; c < 4; ++c) {
        int k0 = 4*c + 2*h;
        b0[c].x = W0[(k0  )*N_CHANNEL + m]; b0[c].y = W0[(k0+1)*N_CHANNEL + m];
        b1[c].x = W1[(k0  )*N_CHANNEL + m]; b1[c].y = W1[(k0+1)*N_CHANNEL + m];
        b2[c].x = W2[(k0  )*N_CHANNEL + m]; b2[c].y = W2[(k0+1)*N_CHANNEL + m];
    }

    // fn0/fn1/fn2 = RBF(16x16) @ W(16x16) via 4 chained K=4 f32 WMMAs each
    v8f f0 = {}, f1 = {}, f2 = {};
    #pragma unroll
    for (int c = 0; c < 4; ++c) {
        f0 = __builtin_amdgcn_wmma_f32_16x16x4_f32(false, a[c], false, b0[c], (short)0, f0, false, false);
        f1 = __builtin_amdgcn_wmma_f32_16x16x4_f32(false, a[c], false, b1[c], (short)0, f1, false, false);
        f2 = __builtin_amdgcn_wmma_f32_16x16x4_f32(false, a[c], false, b2[c], (short)0, f2, false, false);
    }

    // ---- Stage 3: lane (h,m) owns channel m of edges wbase + 8h + r, r=0..7
    #pragma unroll
    for (int r = 0; r < 8; ++r) {
        int  el    = wbase + 8*h + r;
        int  ge    = block_e0 + el;
        bool valid = (ge < n_edges);

        int   ii = s_ii[el], jj = s_jj[el];
        float ux = s_ux[el], uy = s_uy[el], uz = s_uz[el];

        const float* ti = tin + (ii*N_CHANNEL + m)*3;
        const float* tj = tin + (jj*N_CHANNEL + m)*3;
        float tix = ti[0], tiy = ti[1], tiz = ti[2];
        float tjx = tj[0], tjy = tj[1], tjz = tj[2];

        float s0  = tix*tjx + tiy*tjy + tiz*tjz;         // ti . tj
        float cx  = tiy*tjz - tiz*tjy;                    // cross(ti, tj)
        float cy  = tiz*tjx - tix*tjz;
        float cz  = tix*tjy - tiy*tjx;
        float tiu = tix*ux + tiy*uy + tiz*uz;             // ti . u
        float cu  = cx*ux + cy*uy + cz*uz;                // s1 . u

        float g0 = f0[r], g1 = f1[r], g2 = f2[r];
        float au = g1*s0 + g2*cu;                         // coefficient on u
        float mx = g0*cx + au*ux + g1*tiu*tjx;
        float my = g0*cy + au*uy + g1*tiu*tjy;
        float mz = g0*cz + au*uz + g1*tiu*tjz;

        if (valid) {
            float* o = out + (ii*N_CHANNEL + m)*3;
            atomicAdd(o + 0, mx * INV_NORM);
            atomicAdd(o + 1, my * INV_NORM);
            atomicAdd(o + 2, mz * INV_NORM);
        }
    }
}

extern "C" void kernel_launch(void* const* d_in, const int* in_sizes, int n_in,
                              void* d_out, int out_size, void* d_ws, size_t ws_size,
                              hipStream_t stream) {
    const float* tin   = (const float*)d_in[0];
    const float* coord = (const float*)d_in[1];
    const float* W0    = (const float*)d_in[2];
    const float* W1    = (const float*)d_in[3];
    const float* W2    = (const float*)d_in[4];
    const int*   idx_i = (const int*)d_in[5];
    const int*   idx_j = (const int*)d_in[6];
    float*       out   = (float*)d_out;
    int n_edges = in_sizes[5];

    zero_f32_kernel<<<(out_size + 255) / 256, 256, 0, stream>>>(out, out_size);

    int nblocks = (n_edges + EDGES_PER_BLOCK - 1) / EDGES_PER_BLOCK;
    spin_edge_kernel<<<nblocks, THREADS, 0, stream>>>(
        tin, coord, W0, W1, W2, idx_i, idx_j, out, n_edges);
}